// BottomUpHTMM_39926015983659
// MI455X (gfx1250) — compile-verified
//
#include <hip/hip_runtime.h>

// ---------------------------------------------------------------------------
// Bottom-up HTMM on a full quad-tree (L=4, depth=7), G=16 models, C=8 states.
// Level kernels use V_WMMA_F32_16X16X4_F32 (exact fp32 matrix core) for the
// per-level 8x32 contractions; everything else fused per level.
// ---------------------------------------------------------------------------

typedef __attribute__((ext_vector_type(2))) float v2f;
typedef __attribute__((ext_vector_type(8))) float v8f;

#define L_DEG 4
#define C_ST 8
#define N_GEN 16
#define M_LAB 256
#define T_SIZE 21845
#define S_LEAF 5461           // first leaf index
#define N_LEAF 16384
#define GC (N_GEN * C_ST)     // 128: per-node (g,c) slab

__device__ __forceinline__ v8f wmma_f32_4(v2f A, v2f B, v8f C) {
    // D = A(16x4,f32) * B(4x16,f32) + C(16x16,f32)
    return __builtin_amdgcn_wmma_f32_16x16x4_f32(
        /*neg_a=*/false, A, /*neg_b=*/false, B,
        /*c_mod=*/(short)0, C, /*reuse_a=*/false, /*reuse_b=*/false);
}

// ---------------------------------------------------------------------------
// setup1: all softmaxes. idx ranges dispatch the jobs.
//   [0,512):   sm_a over states i, per (g,j,l)        -> SA [g][i][j*4+l]
//   [512,640): sm_b + log over labels, per (g,c)      -> SMB/LSMB [g][c][m]
//   [640,704): sm_pi + log over states, per (g,l)     -> SMPI/LSMPI [g][i][l]
//   [704,720): sm_sp + log over l, per g              -> SMSP/LSMSP [g][l]
//   [720,736): zero the 16 log-likelihood accumulators
// ---------------------------------------------------------------------------
__global__ __launch_bounds__(256) void htmm_setup1(
    const float* __restrict__ a, const float* __restrict__ b,
    const float* __restrict__ pi, const float* __restrict__ sp,
    float* __restrict__ SA, float* __restrict__ SMB, float* __restrict__ LSMB,
    float* __restrict__ SMPI, float* __restrict__ LSMPI,
    float* __restrict__ SMSP, float* __restrict__ LSMSP,
    float* __restrict__ LH)
{
    int idx = blockIdx.x * blockDim.x + threadIdx.x;
    if (idx < 512) {
        int g = idx >> 5, j = (idx & 31) >> 2, l = idx & 3;
        const float* src = a + g * 256 + j * 4 + l;      // stride 32 over i
        float mx = -3.4e38f;
        #pragma unroll
        for (int i = 0; i < 8; ++i) mx = fmaxf(mx, src[i * 32]);
        float ex[8], sum = 0.f;
        #pragma unroll
        for (int i = 0; i < 8; ++i) { ex[i] = expf(src[i * 32] - mx); sum += ex[i]; }
        float inv = 1.0f / sum;
        #pragma unroll
        for (int i = 0; i < 8; ++i) SA[g * 256 + i * 32 + j * 4 + l] = ex[i] * inv;
    } else if (idx < 640) {
        int q = idx - 512, g = q >> 3, c = q & 7;
        const float* src = b + g * 2048 + c * 256;
        float mx = -3.4e38f;
        for (int m = 0; m < 256; ++m) mx = fmaxf(mx, src[m]);
        float sum = 0.f;
        for (int m = 0; m < 256; ++m) sum += expf(src[m] - mx);
        float inv = 1.0f / sum, ls = logf(sum);
        for (int m = 0; m < 256; ++m) {
            float z = src[m] - mx;
            SMB[g * 2048 + c * 256 + m]  = expf(z) * inv;
            LSMB[g * 2048 + c * 256 + m] = z - ls;
        }
    } else if (idx < 704) {
        int q = idx - 640, g = q >> 2, l = q & 3;
        const float* src = pi + g * 32 + l;              // stride 4 over i
        float mx = -3.4e38f;
        #pragma unroll
        for (int i = 0; i < 8; ++i) mx = fmaxf(mx, src[i * 4]);
        float ex[8], sum = 0.f;
        #pragma unroll
        for (int i = 0; i < 8; ++i) { ex[i] = expf(src[i * 4] - mx); sum += ex[i]; }
        float inv = 1.0f / sum, ls = logf(sum);
        #pragma unroll
        for (int i = 0; i < 8; ++i) {
            SMPI[g * 32 + i * 4 + l]  = ex[i] * inv;
            LSMPI[g * 32 + i * 4 + l] = (src[i * 4] - mx) - ls;
        }
    } else if (idx < 720) {
        int g = idx - 704;
        const float* src = sp + g * 4;
        float mx = fmaxf(fmaxf(src[0], src[1]), fmaxf(src[2], src[3]));
        float ex[4], sum = 0.f;
        #pragma unroll
        for (int l = 0; l < 4; ++l) { ex[l] = expf(src[l] - mx); sum += ex[l]; }
        float inv = 1.0f / sum, ls = logf(sum);
        #pragma unroll
        for (int l = 0; l < 4; ++l) {
            SMSP[g * 4 + l]  = ex[l] * inv;
            LSMSP[g * 4 + l] = (src[l] - mx) - ls;
        }
    } else if (idx < 736) {
        LH[idx - 720] = 0.0f;
    }
}

// setup2: a_sp = sm_a * sm_sp[l];  awl = a_sp * log(sm_a)
// Also writes k-major transposed copies ASPT/AWLT [g][k][i] so the downward
// kernel's B operands are contiguous v2f loads.
__global__ __launch_bounds__(256) void htmm_setup2(
    const float* __restrict__ SA, const float* __restrict__ SMSP,
    float* __restrict__ ASP, float* __restrict__ AWL,
    float* __restrict__ ASPT, float* __restrict__ AWLT)
{
    int idx = blockIdx.x * blockDim.x + threadIdx.x;
    if (idx < 4096) {
        int g = idx >> 8, r = idx & 255;
        int i = r >> 5, k = r & 31, l = k & 3;
        float sa = SA[idx];
        float v = sa * SMSP[g * 4 + l];
        float w = v * logf(sa);
        ASP[idx] = v;
        AWL[idx] = w;
        ASPT[g * 256 + k * 8 + i] = v;
        AWLT[g * 256 + k * 8 + i] = w;
    }
}

// ---------------------------------------------------------------------------
// Leaves: prior = sm_pi[:, pos], beta = normalize(prior * sm_b[:, label])
// one thread per (leaf, g)
// ---------------------------------------------------------------------------
__global__ __launch_bounds__(256) void htmm_leaf(
    const int* __restrict__ t, const float* __restrict__ SMPI,
    const float* __restrict__ SMB,
    float* __restrict__ prior, float* __restrict__ beta)
{
    int idx = blockIdx.x * blockDim.x + threadIdx.x;
    if (idx >= N_LEAF * N_GEN) return;
    int u = S_LEAF + (idx >> 4), g = idx & 15;
    int lab = t[u * 7 + 0];
    int pos = t[u * 7 + 2];
    float pv[8], bv[8], sum = 0.f;
    #pragma unroll
    for (int i = 0; i < 8; ++i) {
        pv[i] = SMPI[g * 32 + i * 4 + pos];
        bv[i] = pv[i] * SMB[g * 2048 + i * 256 + lab];
        sum += bv[i];
    }
    float inv = 1.0f / sum;
    int o = u * GC + g * 8;
    #pragma unroll
    for (int i = 0; i < 8; ++i) {
        prior[o + i] = pv[i];
        beta[o + i]  = bv[i] * inv;
    }
}

// ---------------------------------------------------------------------------
// Upward level: one wave per (g, 16-node tile).
// GEMM D[i,p] = sum_k A[i,k] * X[k,p], k = j*4+l (K=32, 8 chunks of 4),
// done twice (prior-children and beta-children), then per-node normalize.
// A rows 8..15 DUPLICATE rows 0..7 (nidx&7): no masking, no exec branches;
// lanes 16..31 of D just hold copies that are never read.
// ---------------------------------------------------------------------------
__global__ __launch_bounds__(256) void htmm_up(
    const int* __restrict__ t, const float* __restrict__ ASP,
    const float* __restrict__ SMB,
    float* __restrict__ prior, float* __restrict__ beta,
    float* __restrict__ beta_il, int s, int e)
{
    int wid  = (blockIdx.x * blockDim.x + threadIdx.x) >> 5;
    int lane = threadIdx.x & 31;
    int nTiles = (e - s + 15) >> 4;
    if (wid >= nTiles * N_GEN) return;
    int g    = wid & 15;
    int tile = wid >> 4;
    int hf   = lane >> 4;        // which K pair within a chunk
    int nidx = lane & 15;        // A row / B node column
    int node = s + tile * 16 + nidx;
    int ncl  = (node < e) ? node : s;

    v8f accP = {0.f,0.f,0.f,0.f,0.f,0.f,0.f,0.f};
    v8f accB = {0.f,0.f,0.f,0.f,0.f,0.f,0.f,0.f};
    // row-duplicated A: row m -> a_sp row (m & 7); contiguous v2f per chunk
    const v2f* Aall = (const v2f*)(ASP + g * 256 + (nidx & 7) * 32);

    #pragma unroll
    for (int c = 0; c < 8; ++c) {              // chunk c <=> j = c
        v2f A = Aall[c * 2 + hf];              // k0 = c*4 + hf*2
        int ch0 = 4 * ncl + 1 + hf * 2;        // children l = 2*hf, 2*hf+1
        int b0  = ch0 * GC + g * 8 + c;
        v2f Bp, Bb;
        Bp.x = prior[b0]; Bp.y = prior[b0 + GC];
        Bb.x = beta[b0];  Bb.y = beta[b0 + GC];
        accP = wmma_f32_4(A, Bp, accP);
        accB = wmma_f32_4(A, Bb, accB);
    }

    // lanes 0..15 hold all 8 states for their node in acc[0..7]
    if (hf == 0 && node < e) {
        int lab = t[node * 7 + 0];
        const float* bg = SMB + g * 2048 + lab;
        float tmp[8], sum = 0.f;
        #pragma unroll
        for (int r = 0; r < 8; ++r) {          // tmp = emis * num_beta
            tmp[r] = bg[r * 256] * accB[r];
            sum += tmp[r];
        }
        float inv = 1.0f / sum;
        int o = node * GC + g * 8;
        #pragma unroll
        for (int r = 0; r < 8; ++r) {
            prior[o + r]   = accP[r];
            beta_il[o + r] = accB[r] / accP[r];
            beta[o + r]    = tmp[r] * inv;
        }
    }
}

// eps_i[0] = beta[0]
__global__ void htmm_root(const float* __restrict__ beta, float* __restrict__ eps)
{
    int i = threadIdx.x;
    if (i < GC) eps[i] = beta[i];
}

// ---------------------------------------------------------------------------
// Downward level: one wave per (g, 16-node tile).
// D[m,k]  = sum_i pe[m,i] * a_sp[i,k]   (pe = eps/(prior*beta_il))
// D2[m,k] = sum_i pe[m,i] * awl[i,k]    (awl = a_sp*log sm_a)
// res = D * beta_child; scatter res -> eps of children;
// lh_g += D2*beta_child  (a_lh)  +  res*log_sp[l]  (sp_lh)
// B operands come from the k-major transposed copies -> contiguous v2f loads.
// ---------------------------------------------------------------------------
__global__ __launch_bounds__(256) void htmm_down(
    const float* __restrict__ ASPT, const float* __restrict__ AWLT,
    const float* __restrict__ prior, const float* __restrict__ beta,
    const float* __restrict__ beta_il, float* __restrict__ eps,
    const float* __restrict__ LSMSP, float* __restrict__ LH,
    int s, int e)
{
    int wid  = (blockIdx.x * blockDim.x + threadIdx.x) >> 5;
    int lane = threadIdx.x & 31;
    int nTiles = (e - s + 15) >> 4;
    if (wid >= nTiles * N_GEN) return;
    int g    = wid & 15;
    int tile = wid >> 4;
    int hf   = lane >> 4;
    int nidx = lane & 15;
    int nodeA = s + tile * 16 + nidx;          // node for this lane's A row
    int nclA  = (nodeA < e) ? nodeA : s;
    int pb    = nclA * GC + g * 8;

    v8f D0 = {0.f,0.f,0.f,0.f,0.f,0.f,0.f,0.f};
    v8f D1 = D0, E0 = D0, E1 = D0;
    const float* mAT = ASPT + g * 256;         // [k][i] layout
    const float* mWT = AWLT + g * 256;

    #pragma unroll
    for (int c = 0; c < 2; ++c) {              // K = i (8), chunks of 4
        int i0 = c * 4 + hf * 2;
        v2f ev = *(const v2f*)(eps     + pb + i0);
        v2f pv = *(const v2f*)(prior   + pb + i0);
        v2f bv = *(const v2f*)(beta_il + pb + i0);
        v2f A  = ev / (pv * bv);
        int k0 = nidx, k1 = 16 + nidx;
        v2f B0a = *(const v2f*)(mAT + k0 * 8 + i0);
        v2f B1a = *(const v2f*)(mAT + k1 * 8 + i0);
        v2f B0w = *(const v2f*)(mWT + k0 * 8 + i0);
        v2f B1w = *(const v2f*)(mWT + k1 * 8 + i0);
        D0 = wmma_f32_4(A, B0a, D0);
        D1 = wmma_f32_4(A, B1a, D1);
        E0 = wmma_f32_4(A, B0w, E0);
        E1 = wmma_f32_4(A, B1w, E1);
    }

    float acc = 0.0f;
    #pragma unroll
    for (int t2 = 0; t2 < 2; ++t2) {
        int k = t2 * 16 + nidx;
        int j = k >> 2, l = k & 3;
        float lsp = LSMSP[g * 4 + l];
        v8f D = t2 ? D1 : D0;
        v8f E = t2 ? E1 : E0;
        #pragma unroll
        for (int r = 0; r < 8; ++r) {
            int m = r + 8 * hf;                // D row -> node in tile
            int node = s + tile * 16 + m;
            if (node < e) {
                int child = 4 * node + 1 + l;
                int co = child * GC + g * 8 + j;
                float bch = beta[co];
                float res = D[r] * bch;
                eps[co] = res;                 // unique (child,g,j) writer
                acc += E[r] * bch + res * lsp;
            }
        }
    }
    atomicAdd(&LH[g], acc);
}

// ---------------------------------------------------------------------------
// Final b_lh (all nodes) + pi_lh (leaves): one thread per (node, g).
// Butterfly over lane^16 (same g, adjacent node) halves the atomics.
// ---------------------------------------------------------------------------
__global__ __launch_bounds__(256) void htmm_lh(
    const int* __restrict__ t, const float* __restrict__ eps,
    const float* __restrict__ LSMB, const float* __restrict__ LSMPI,
    float* __restrict__ LH)
{
    int idx = blockIdx.x * blockDim.x + threadIdx.x;
    float val = 0.0f;
    if (idx < T_SIZE * N_GEN) {
        int u = idx >> 4, g = idx & 15;
        int lab = t[u * 7 + 0];
        int pos = t[u * 7 + 2];
        const float* ev = eps + u * GC + g * 8;
        const float* lb = LSMB + g * 2048 + lab;
        #pragma unroll
        for (int i = 0; i < 8; ++i) val += ev[i] * lb[i * 256];
        if (u >= S_LEAF) {
            const float* lp = LSMPI + g * 32 + pos;
            #pragma unroll
            for (int i = 0; i < 8; ++i) val += ev[i] * lp[i * 4];
        }
    }
    val += __shfl_xor(val, 16, 32);
    if ((threadIdx.x & 31) < 16)
        atomicAdd(&LH[idx & 15], val);
}

__global__ void htmm_out(const float* __restrict__ LH, float* __restrict__ out)
{
    int g = threadIdx.x;
    if (g < N_GEN) out[g] = LH[g];
}

// ---------------------------------------------------------------------------
extern "C" void kernel_launch(void* const* d_in, const int* in_sizes, int n_in,
                              void* d_out, int out_size, void* d_ws, size_t ws_size,
                              hipStream_t stream)
{
    (void)in_sizes; (void)n_in; (void)out_size; (void)ws_size;
    const int*   t  = (const int*)  d_in[0];   // (21845,7) int32
    const float* a  = (const float*)d_in[2];   // (16,8,8,4)
    const float* b  = (const float*)d_in[3];   // (16,8,256)
    const float* pi = (const float*)d_in[4];   // (16,8,4)
    const float* sp = (const float*)d_in[5];   // (16,4)
    float* out = (float*)d_out;                // (16,)

    float* w = (float*)d_ws;
    const int NPG = T_SIZE * GC;               // 2,796,160 floats per array
    float* prior   = w;
    float* beta    = w + (size_t)NPG;
    float* beta_il = w + (size_t)2 * NPG;
    float* eps     = w + (size_t)3 * NPG;
    float* SA      = w + (size_t)4 * NPG;      // sm_a              (4096)
    float* ASP     = SA + 4096;                // a_sp              (4096)
    float* AWL     = ASP + 4096;               // a_sp*log sm_a     (4096)
    float* ASPT    = AWL + 4096;               // a_sp   [g][k][i]  (4096)
    float* AWLT    = ASPT + 4096;              // awl    [g][k][i]  (4096)
    float* SMB     = AWLT + 4096;              // sm_b              (32768)
    float* LSMB    = SMB + 32768;              // log sm_b          (32768)
    float* SMPI    = LSMB + 32768;             // sm_pi             (512)
    float* LSMPI   = SMPI + 512;               // log sm_pi         (512)
    float* SMSP    = LSMPI + 512;              // sm_sp             (64)
    float* LSMSP   = SMSP + 64;                // log sm_sp         (64)
    float* LH      = LSMSP + 64;               // per-g accumulators (16)

    static const int LIM[9] = {0, 1, 5, 21, 85, 341, 1365, 5461, 21845};

    htmm_setup1<<<3, 256, 0, stream>>>(a, b, pi, sp, SA, SMB, LSMB,
                                       SMPI, LSMPI, SMSP, LSMSP, LH);
    htmm_setup2<<<16, 256, 0, stream>>>(SA, SMSP, ASP, AWL, ASPT, AWLT);

    htmm_leaf<<<(N_LEAF * N_GEN + 255) / 256, 256, 0, stream>>>(
        t, SMPI, SMB, prior, beta);

    for (int lev = 6; lev >= 0; --lev) {
        int s = LIM[lev], e = LIM[lev + 1];
        int tiles  = (e - s + 15) / 16;
        int waves  = tiles * N_GEN;
        int blocks = (waves + 7) / 8;          // 8 waves / 256-thread block
        htmm_up<<<blocks, 256, 0, stream>>>(t, ASP, SMB, prior, beta, beta_il, s, e);
    }

    htmm_root<<<1, 128, 0, stream>>>(beta, eps);

    for (int lev = 0; lev <= 6; ++lev) {
        int s = LIM[lev], e = LIM[lev + 1];
        int tiles  = (e - s + 15) / 16;
        int waves  = tiles * N_GEN;
        int blocks = (waves + 7) / 8;
        htmm_down<<<blocks, 256, 0, stream>>>(ASPT, AWLT, prior, beta, beta_il,
                                              eps, LSMSP, LH, s, e);
    }

    htmm_lh<<<(T_SIZE * N_GEN + 255) / 256, 256, 0, stream>>>(
        t, eps, LSMB, LSMPI, LH);
    htmm_out<<<1, 32, 0, stream>>>(LH, out);
}